// Tracker_61074434949826
// MI455X (gfx1250) — compile-verified
//
#include <hip/hip_runtime.h>
#include <cstdint>
#include <cstddef>

// Problem geometry (from the reference):
//   GRID_H = 67, GRID_W = 90, N = 6030 elements per batch, B = 4096
//   grid coords: gx = col*7+7, gy = row*7+7 ; mask: dr^2+dc^2 <= 25 (grid cells)
#define GH 67
#define GW 90
#define NPB (GH * GW)       // 6030
#define NT 256              // 8 wave32s
#define NW (NT / 32)
#define NCHUNK (NPB / 2)    // 3015 x b64 async chunks == exactly 6030 floats

// ---- CDNA5 async global->LDS copy (gfx1250, ASYNCcnt-tracked), GVS form ----
// vdst  = VGPR with 32-bit LDS byte address (low 32 bits of flat LDS pointer,
//         see ISA 10.2)
// vaddr = 32-bit per-lane byte offset, saddr = uniform 64-bit base in SGPRs
__device__ __forceinline__ void async_g2l_b64(const float* base, unsigned byte_off,
                                              const float* l) {
  unsigned loff = (unsigned)(uintptr_t)l;
  asm volatile("global_load_async_to_lds_b64 %0, %1, %2"
               :: "v"(loff), "v"(byte_off), "s"(base) : "memory");
}
__device__ __forceinline__ void wait_async0() {
  asm volatile("s_wait_asynccnt 0" ::: "memory");
}

__global__ __launch_bounds__(NT)
void tracker_kernel(const float* __restrict__ cost, float* __restrict__ out) {
  __shared__ float sdata[NPB];       // 24120 B staged heatmap
  __shared__ float swval[NW];
  __shared__ int   swidx[NW];
  __shared__ float sred[NW][8];
  __shared__ float sgmax;
  __shared__ int   sgidx;

  const int b    = blockIdx.x;
  const int tid  = threadIdx.x;
  const int lane = tid & 31;
  const int wave = tid >> 5;
  const float* src = cost + (size_t)b * NPB;   // block-uniform -> lives in SGPRs

  // ---- stage heatmap into LDS via async DMA-style loads (GVS addressing) ----
  // 8 bytes/lane * 32 lanes = 256 contiguous bytes per issued instruction.
  for (int c = tid; c < NCHUNK; c += NT)
    async_g2l_b64(src, (unsigned)(8u * c), &sdata[2 * c]);
  wait_async0();
  __syncthreads();

  // ---- pass 1: max + first-occurrence argmax ----
  float bv = -3.402823466e38f; int bi = NPB;
  for (int i = tid; i < NPB; i += NT) {
    float v = sdata[i];
    if (v > bv) { bv = v; bi = i; }        // strictly-greater => earliest index kept
  }
  #pragma unroll
  for (int off = 16; off > 0; off >>= 1) {
    float ov = __shfl_xor(bv, off);
    int   oi = __shfl_xor(bi, off);
    if (ov > bv || (ov == bv && oi < bi)) { bv = ov; bi = oi; }
  }
  if (lane == 0) { swval[wave] = bv; swidx[wave] = bi; }
  __syncthreads();
  if (tid == 0) {
    float gv = swval[0]; int gi = swidx[0];
    #pragma unroll
    for (int w = 1; w < NW; ++w)
      if (swval[w] > gv || (swval[w] == gv && swidx[w] < gi)) { gv = swval[w]; gi = swidx[w]; }
    sgmax = gv; sgidx = gi;
  }
  __syncthreads();
  const float gmax = sgmax;
  const int   gidx = sgidx;
  const int   arow = gidx / GW;
  const int   acol = gidx - arow * GW;

  // ---- pass 2: softmax-Z + masked moments (LDS re-read, zero extra HBM) ----
  float Z = 0.f, m = 0.f, sx = 0.f, sy = 0.f, cnt = 0.f, ux = 0.f, uy = 0.f;
  for (int i = tid; i < NPB; i += NT) {
    float e = __expf(sdata[i] - gmax);
    Z += e;
    int r  = i / GW;
    int c  = i - r * GW;
    int dr = r - arow, dc = c - acol;
    if (dr * dr + dc * dc <= 25) {
      float gx = (float)(c * 7 + 7);
      float gy = (float)(r * 7 + 7);
      m   += e;
      sx  += gx * e;
      sy  += gy * e;
      cnt += 1.0f;
      ux  += gx;       // only needed for the sum<EPS uniform branch
      uy  += gy;
    }
  }
  float acc[7] = {Z, m, sx, sy, cnt, ux, uy};
  #pragma unroll
  for (int off = 16; off > 0; off >>= 1)
    #pragma unroll
    for (int k = 0; k < 7; ++k) acc[k] += __shfl_xor(acc[k], off);
  if (lane == 0)
    #pragma unroll
    for (int k = 0; k < 7; ++k) sred[wave][k] = acc[k];
  __syncthreads();

  if (tid == 0) {
    float t[7];
    #pragma unroll
    for (int k = 0; k < 7; ++k) {
      float s = 0.f;
      #pragma unroll
      for (int w = 0; w < NW; ++w) s += sred[w][k];
      t[k] = s;
    }
    const float Zt = t[0], mt = t[1], sxt = t[2], syt = t[3];
    const float cm = t[4], uxt = t[5], uyt = t[6];
    const float hm_sum = mt / Zt;          // sum of masked softmax
    float px, py;
    if (hm_sum < 1e-8f) {
      // hm_final = sm*mask + (1/cnt)*mask : denom = hm_sum + 1
      const float uw    = 1.0f / cm;
      const float denom = hm_sum + 1.0f;
      px = (sxt / Zt + uw * uxt) / denom;
      py = (syt / Zt + uw * uyt) / denom;
    } else {
      px = sxt / mt;                       // Z cancels
      py = syt / mt;
    }
    out[2 * b + 0] = 2.0f * (px / 639.0f) - 1.0f;   // VIDEO_W-1
    out[2 * b + 1] = 2.0f * (py / 479.0f) - 1.0f;   // VIDEO_H-1
  }
}

extern "C" void kernel_launch(void* const* d_in, const int* in_sizes, int n_in,
                              void* d_out, int out_size, void* d_ws, size_t ws_size,
                              hipStream_t stream) {
  const float* cost = (const float*)d_in[0];
  float* out = (float*)d_out;
  const int B = in_sizes[0] / NPB;       // 4096
  tracker_kernel<<<dim3(B), dim3(NT), 0, stream>>>(cost, out);
}